// NeuralRenderer_13125420056827
// MI455X (gfx1250) — compile-verified
//
#include <hip/hip_runtime.h>
#include <math.h>

// Problem constants (match reference)
#define BSZ 2
#define NV 3456
#define FC 6912          // multiple of 16 and of 64 -> no padding, all faces valid
#define IMG 256
#define OFFSET_Z 5.0f
#define EPSA 1e-9f
#define BIG 1e9f
// exp(mb/sigma) = exp2(mb * (1/sigma) * log2(e));  1/sigma = 1e4
#define K_SCALE 14426.950408889634f
#define NEG_HUGE -1.0e30f

typedef __attribute__((ext_vector_type(2))) float v2f;
typedef __attribute__((ext_vector_type(8))) float v8f;

// ---------------------------------------------------------------------------
// Kernel 1: quaternion rotate + scale/translate + y-flip + z-offset
// ---------------------------------------------------------------------------
__global__ void xform_verts(const float* __restrict__ vin,
                            const float* __restrict__ cams,
                            float* __restrict__ vout) {
  int i = blockIdx.x * blockDim.x + threadIdx.x;
  if (i >= BSZ * NV) return;
  int b = i / NV;
  const float* cam = cams + b * 7;
  float s  = cam[0], tx = cam[1], ty = cam[2];
  float qw = cam[3], qx = cam[4], qy = cam[5], qz = cam[6];
  float X = vin[i*3+0], Y = vin[i*3+1], Z = vin[i*3+2];
  // t = H((0,X,Y,Z), conj(q))
  float tw = X*qx + Y*qy + Z*qz;
  float ta = X*qw - Y*qz + Z*qy;
  float tb = X*qz + Y*qw - Z*qx;
  float tc = -X*qy + Y*qx + Z*qw;
  // r = vector part of H(q, t)
  float rx = qw*ta + qx*tw + qy*tc - qz*tb;
  float ry = qw*tb - qx*tc + qy*tw + qz*ta;
  float rz = qw*tc + qx*tb - qy*ta + qz*tw;
  vout[i*3+0] = s*rx + tx;
  vout[i*3+1] = -(s*ry + ty);       // y-flip folded in
  vout[i*3+2] = s*rz + OFFSET_Z;
}

// ---------------------------------------------------------------------------
// Kernel 2: per-face affine-plane coefficients for lambda0/1/2 and depth.
// Plane-major float4 layout: P[p][b*FC+f] = (a,b,c, p==0 ? ok : 0)
// ---------------------------------------------------------------------------
__global__ void face_coeffs(const int* __restrict__ faces,
                            const float* __restrict__ verts,
                            float4* __restrict__ P) {
  int i = blockIdx.x * blockDim.x + threadIdx.x;
  if (i >= BSZ * FC) return;
  int b = i / FC;
  const float* vb = verts + b * NV * 3;
  int i0 = faces[i*3+0], i1 = faces[i*3+1], i2 = faces[i*3+2];
  float x0 = vb[i0*3], y0 = vb[i0*3+1], z0 = vb[i0*3+2];
  float x1 = vb[i1*3], y1 = vb[i1*3+1], z1 = vb[i1*3+2];
  float x2 = vb[i2*3], y2 = vb[i2*3+1], z2 = vb[i2*3+2];

  float area = (x1-x0)*(y2-y0) - (y1-y0)*(x2-x0);
  bool okb = fabsf(area) > EPSA;
  float inv = 1.0f / (okb ? area : EPSA);
  // lambda0 = e1/area :  e1 = (x2-x1)(py-y1) - (y2-y1)(px-x1)
  float a0 = -(y2-y1)*inv, b0 = (x2-x1)*inv, c0 = ((y2-y1)*x1 - (x2-x1)*y1)*inv;
  // lambda1 = e2/area :  e2 = (x0-x2)(py-y2) - (y0-y2)(px-x2)
  float a1 = -(y0-y2)*inv, b1 = (x0-x2)*inv, c1 = ((y0-y2)*x2 - (x0-x2)*y2)*inv;
  // lambda2 = e0/area :  e0 = (x1-x0)(py-y0) - (y1-y0)(px-x0)
  float a2 = -(y1-y0)*inv, b2 = (x1-x0)*inv, c2 = ((y1-y0)*x0 - (x1-x0)*y0)*inv;
  // depth plane = sum lambda_i * z_i
  float ad = a0*z0 + a1*z1 + a2*z2;
  float bd = b0*z0 + b1*z1 + b2*z2;
  float cd = c0*z0 + c1*z1 + c2*z2;

  const int n = BSZ * FC;
  P[0*n + i] = make_float4(a0, b0, c0, okb ? 1.0f : 0.0f);
  P[1*n + i] = make_float4(a1, b1, c1, 0.0f);
  P[2*n + i] = make_float4(a2, b2, c2, 0.0f);
  P[3*n + i] = make_float4(ad, bd, cd, 0.0f);
}

// ---------------------------------------------------------------------------
// Kernel 3: rasterizer. One wave = 16 consecutive x-pixels of one row,
// looping over all faces 16 at a time via V_WMMA_F32_16X16X4_F32.
//   A (16x4): row m = [px_m, py, 1, 0]   (lanes 0-15: K0,K1; 16-31: K2,K3)
//   B (4x16): col n = [a_n, b_n, c_n, 0] (lanes 0-15: K0,K1; 16-31: K2,K3)
//   D (16x16): lane L holds face n=L&15, pixels m = r (+8 for hi half)
// ---------------------------------------------------------------------------
__global__ void __launch_bounds__(256) raster(const float4* __restrict__ P,
                                              float* __restrict__ out_mask,
                                              int* __restrict__ out_fidx) {
  const int lane = threadIdx.x & 31;
  const int wave = blockIdx.x * (blockDim.x >> 5) + (threadIdx.x >> 5);
  const int b  = wave / (IMG * (IMG / 16));
  const int rm = wave % (IMG * (IMG / 16));
  const int y  = rm / (IMG / 16);
  const int xb = rm % (IMG / 16);
  const bool hi = lane >= 16;
  const int nl = lane & 15;

  // A operand (constant across the whole face loop)
  v2f aop;
  if (!hi) {
    aop.x = ((float)(xb * 16 + lane) + 0.5f) * (2.0f / IMG) - 1.0f;  // px
    aop.y = ((float)y + 0.5f) * (2.0f / IMG) - 1.0f;                 // py
  } else {
    aop.x = 1.0f;   // multiplies c coefficient
    aop.y = 0.0f;   // K=3 unused
  }

  float prod[8], zmin[8];
  int fid[8];
#pragma unroll
  for (int r = 0; r < 8; ++r) { prod[r] = 1.0f; zmin[r] = BIG; fid[r] = -1; }

  const int n = BSZ * FC;
  const int base = b * FC;
  const v8f czero = {};

  for (int f0 = 0; f0 < FC; f0 += 16) {
    const int fi = base + f0 + nl;
    float4 c0 = P[0*n + fi];
    float4 c1 = P[1*n + fi];
    float4 c2 = P[2*n + fi];
    float4 c3 = P[3*n + fi];
    if (f0 + 16 < FC) {    // stream next block toward L0/L2
      __builtin_prefetch((const void*)(P + 0*n + fi + 16), 0, 1);
      __builtin_prefetch((const void*)(P + 1*n + fi + 16), 0, 1);
      __builtin_prefetch((const void*)(P + 2*n + fi + 16), 0, 1);
      __builtin_prefetch((const void*)(P + 3*n + fi + 16), 0, 1);
    }

    v2f b0, b1, b2, b3;
    b0.x = hi ? c0.z : c0.x;  b0.y = hi ? c0.w : c0.y;  // K3 slot x 0 in A
    b1.x = hi ? c1.z : c1.x;  b1.y = hi ? c1.w : c1.y;
    b2.x = hi ? c2.z : c2.x;  b2.y = hi ? c2.w : c2.y;
    b3.x = hi ? c3.z : c3.x;  b3.y = hi ? c3.w : c3.y;

    v8f D0 = __builtin_amdgcn_wmma_f32_16x16x4_f32(false, aop, false, b0, (short)0, czero, false, false);
    v8f D1 = __builtin_amdgcn_wmma_f32_16x16x4_f32(false, aop, false, b1, (short)0, czero, false, false);
    v8f D2 = __builtin_amdgcn_wmma_f32_16x16x4_f32(false, aop, false, b2, (short)0, czero, false, false);
    v8f D3 = __builtin_amdgcn_wmma_f32_16x16x4_f32(false, aop, false, b3, (short)0, czero, false, false);

    const bool ok = c0.w > 0.0f;     // lane-uniform per face (lane owns face f0+nl)
    const int fcur = f0 + nl;
#pragma unroll
    for (int r = 0; r < 8; ++r) {
      float mb = fminf(D0[r], fminf(D1[r], D2[r]));
      // Fold 'ok' into the argument: !ok -> -huge -> exp2->0 -> omc=1,
      // and (mbs >= 0) == (ok && mb >= 0).  All selects, no branches.
      float mbs = ok ? (mb * K_SCALE) : NEG_HUGE;
      // 1 - sigmoid(mb/sigma) = 1/(1+exp(mb/sigma)); raw v_exp_f32 + v_rcp_f32
      float t   = __builtin_amdgcn_exp2f(mbs);
      float omc = __builtin_amdgcn_rcpf(1.0f + t);   // rcp(inf)=0, rcp(1)=1
      prod[r] *= omc;
      float d = (mbs >= 0.0f) ? D3[r] : BIG;
      bool lt = d < zmin[r];                          // first-min-wins
      zmin[r] = lt ? d : zmin[r];
      fid[r]  = lt ? fcur : fid[r];
    }
  }

  // Butterfly reduction over the 16 faces spread across each 16-lane half
  // (xor 1,2,4,8 never crosses the half boundary).
#pragma unroll
  for (int r = 0; r < 8; ++r) {
    float p = prod[r]; float z = zmin[r]; int f = fid[r];
    for (int off = 1; off < 16; off <<= 1) {
      p *= __shfl_xor(p, off, 32);
      float oz = __shfl_xor(z, off, 32);
      int   of = __shfl_xor(f, off, 32);
      if (oz < z || (oz == z && of < f)) { z = oz; f = of; }
    }
    prod[r] = p; zmin[r] = z; fid[r] = f;
  }

  if (nl == 0) {  // lane 0 writes pixels m=0..7, lane 16 writes m=8..15
    const int mbase = (b * IMG + y) * IMG + xb * 16 + (hi ? 8 : 0);
#pragma unroll
    for (int r = 0; r < 8; ++r) {
      out_mask[mbase + r] = 1.0f - prod[r];
      out_fidx[mbase + r] = fid[r];
    }
  }
}

// ---------------------------------------------------------------------------
extern "C" void kernel_launch(void* const* d_in, const int* in_sizes, int n_in,
                              void* d_out, int out_size, void* d_ws, size_t ws_size,
                              hipStream_t stream) {
  const float* vertices = (const float*)d_in[0];   // (B,N,3) f32
  const float* cams     = (const float*)d_in[1];   // (B,7)   f32
  const int*   faces    = (const int*)d_in[2];     // (B,F,3) i32

  float* ws      = (float*)d_ws;
  float* verts_t = ws;                              // BSZ*NV*3 floats (16B-aligned block size)
  float4* coeff  = (float4*)(ws + BSZ * NV * 3);    // 4 planes * BSZ*FC float4s

  float* out_mask = (float*)d_out;                  // B*IMG*IMG floats
  int*   out_fidx = (int*)d_out + BSZ * IMG * IMG;  // B*IMG*IMG int32 (same width)

  xform_verts<<<(BSZ * NV + 255) / 256, 256, 0, stream>>>(vertices, cams, verts_t);
  face_coeffs<<<(BSZ * FC + 255) / 256, 256, 0, stream>>>(faces, verts_t, coeff);

  const int waves  = BSZ * IMG * (IMG / 16);        // 8192 waves
  const int blocks = waves / 8;                     // 256 threads = 8 wave32 per block
  raster<<<blocks, 256, 0, stream>>>((const float4*)coeff, out_mask, out_fidx);
}